// GraphProp_10892037063246
// MI455X (gfx1250) — compile-verified
//
#include <hip/hip_runtime.h>
#include <hip/hip_bf16.h>

// ---------------------------------------------------------------------------
// GNN message passing on MI455X (gfx1250), fp32 WMMA + async-LDS gather path.
//
// Roofline: edge GEMM = 23.6 GFLOP/round fp32 dominates; hv (5.1MB), a
// (10.2MB), all weights fit in 192MB L2, so gathers + scatter atomics are
// L2-resident. HBM traffic ~80MB/round (~4us @ 23.3TB/s) -> compute/L2 bound.
// Reference is fp32 -> V_WMMA_F32_16X16X4_F32. Edge-tile gathers use
// GLOBAL_LOAD_ASYNC_TO_LDS_B128 (ASYNCcnt) with per-wave double buffering so
// the L2 gather latency hides behind the 288 WMMAs of the previous tile.
// ---------------------------------------------------------------------------

#define NNODES 20000
#define NEDGES 640000
#define HDIM   64
#define EFEAT  16
#define KMSG   (2*HDIM + EFEAT)   // 144
#define NMSG   (2*HDIM)           // 128
#define H3     (3*HDIM)           // 192

typedef float v2f __attribute__((ext_vector_type(2)));
typedef float v4f __attribute__((ext_vector_type(4)));
typedef float v8f __attribute__((ext_vector_type(8)));

// D = A(16x4) * B(4x16) + C(16x16), fp32, wave32.
// A layout (ISA 7.12.2): lane L holds row M=L&15; elems = K {0,1} (lanes 0-15)
// or K {2,3} (lanes 16-31). B symmetric: lane L holds col N=L&15 with same K
// split. C/D: VGPR r -> rows r (lanes 0-15) / r+8 (lanes 16-31), N=L&15.
__device__ __forceinline__ v8f wmma4_f32(v2f a, v2f b, v8f c) {
  return __builtin_amdgcn_wmma_f32_16x16x4_f32(false, a, false, b,
                                               (short)0, c, false, false);
}

// Async copy 16B global -> LDS (per-lane addresses), tracked by ASYNCcnt.
__device__ __forceinline__ void async_g2l_b128(void* lds_ptr, const void* gptr) {
  unsigned lds_off = (unsigned)(uintptr_t)lds_ptr;   // low 32b of flat = LDS off
  unsigned long long ga = (unsigned long long)(uintptr_t)gptr;
  asm volatile("global_load_async_to_lds_b128 %0, %1, off"
               :: "v"(lds_off), "v"(ga) : "memory");
}

__device__ __forceinline__ void wait_async_le18() {
  asm volatile("s_wait_asynccnt 18" ::: "memory");
}
__device__ __forceinline__ void wave_lds_fence() {
  asm volatile("s_wait_dscnt 0" ::: "memory");
}

// ---------------------------------------------------------------------------
__global__ void zero_kernel(float* __restrict__ p, int n) {
  for (int i = blockIdx.x * blockDim.x + threadIdx.x; i < n;
       i += gridDim.x * blockDim.x)
    p[i] = 0.0f;
}

// ---------------------------------------------------------------------------
// Edge stage: msgs = concat(hv[dst], hv[src], e) @ W_msg + b_msg, then
// atomic scatter-add into a[dst]. One wave per 16-edge tile, double-buffered
// async gather staging; W_msg^T staged in LDS once per block.
#define EDGE_WAVES 4
#define SIN_STRIDE 148   // 144+4 pad: conflict-free, rows 16B-aligned (592B)
#define SW_STRIDE  148

__global__ __launch_bounds__(128) void edge_msg_kernel(
    const float* __restrict__ hv, const float* __restrict__ e,
    const int* __restrict__ src, const int* __restrict__ dst,
    const float* __restrict__ Wm, const float* __restrict__ bm,
    float* __restrict__ a)
{
  __shared__ float sWt[NMSG * SW_STRIDE];                  // W^T: [col][k]
  __shared__ float sIn[EDGE_WAVES * 2 * 16 * SIN_STRIDE];  // 2 bufs per wave

  const int tid = threadIdx.x;
  for (int idx = tid; idx < KMSG * NMSG; idx += 128) {
    int k = idx >> 7;        // row of W (K index)
    int c = idx & 127;       // col of W (N index)
    sWt[c * SW_STRIDE + k] = Wm[idx];
  }
  __syncthreads();

  const int lane = tid & 31;
  const int wave = tid >> 5;
  const int hi   = lane >> 4;    // lane-half: selects K {0,1} vs {2,3}
  const int lr   = lane & 15;
  float* buf[2] = { &sIn[(wave * 2 + 0) * 16 * SIN_STRIDE],
                    &sIn[(wave * 2 + 1) * 16 * SIN_STRIDE] };

  const int nTiles     = NEDGES / 16;               // 40000, exact
  const int waveGid    = blockIdx.x * EDGE_WAVES + wave;
  const int waveStride = gridDim.x * EDGE_WAVES;

  const v8f vzero = {0.f,0.f,0.f,0.f,0.f,0.f,0.f,0.f};

  // --- async gather of one [16 x 144] tile into an LDS buffer: 18 b128s ----
  auto stage_async = [&](int tile, float* sbuf) {
    const int ebase = tile * 16;
    // cols 0-63 = hv[dst], 64-127 = hv[src], 128-143 = e; 576 float4 total.
    for (int j = lane; j < 16 * 36; j += 32) {   // 18 iters, no divergence
      int row = j / 36;
      int q   = j - row * 36;
      const float* gp;
      if (q < 16)      gp = hv + dst[ebase + row] * HDIM + q * 4;
      else if (q < 32) gp = hv + src[ebase + row] * HDIM + (q - 16) * 4;
      else             gp = e + (ebase + row) * EFEAT + (q - 32) * 4;
      async_g2l_b128(sbuf + row * SIN_STRIDE + q * 4, gp);
    }
  };

  int tile = waveGid;
  if (tile < nTiles) {
    stage_async(tile, buf[0]);
    int cur = 0;
    for (; tile < nTiles; tile += waveStride) {
      // WAR: previous compute's ds reads on buf[cur^1] must finish before the
      // async engine overwrites it.
      wave_lds_fence();
      int next = tile + waveStride;
      stage_async(next < nTiles ? next : tile, buf[cur ^ 1]);  // always 18
      // Async loads complete in order: <=18 outstanding => buf[cur] resident.
      wait_async_le18();

      float* sA = buf[cur];
      v8f acc[8];
#pragma unroll
      for (int nt = 0; nt < 8; ++nt) acc[nt] = vzero;

#pragma unroll 4
      for (int kk = 0; kk < KMSG / 4; ++kk) {          // 36 K-steps
        const int kofs = kk * 4 + hi * 2;
        v2f afrag = *(const v2f*)(sA + lr * SIN_STRIDE + kofs);
#pragma unroll
        for (int nt = 0; nt < 8; ++nt) {
          v2f bfrag = *(const v2f*)(&sWt[(nt * 16 + lr) * SW_STRIDE + kofs]);
          acc[nt] = wmma4_f32(afrag, bfrag, acc[nt]);
        }
      }

      // Epilogue: +bias, scatter-add by dst (hardware global_atomic_add_f32).
      const int ebase = tile * 16;
      int nodeOff[8];
#pragma unroll
      for (int r = 0; r < 8; ++r)
        nodeOff[r] = dst[ebase + r + hi * 8] * NMSG;
#pragma unroll
      for (int nt = 0; nt < 8; ++nt) {
        const int   col  = nt * 16 + lr;
        const float bias = bm[col];
#pragma unroll
        for (int r = 0; r < 8; ++r)
          unsafeAtomicAdd(a + nodeOff[r] + col, acc[nt][r] + bias);
      }
      cur ^= 1;
      asm volatile("" ::: "memory");
    }
  }
  // Trailing dummy asyncs are drained by S_ENDPGM's implicit wait-idle.
}

// ---------------------------------------------------------------------------
// GRU stage: per 16-node tile compute gi = a@W_ih^T + b_ih (K=128) and
// gh = hv@W_hh^T + b_hh (K=64) with WMMA into LDS, then gate math.
#define GRU_WAVES 4
#define SG_STRIDE 196   // 192+4 pad: lanes 0-15 vs 16-31 land 32 banks apart

__global__ __launch_bounds__(128) void gru_kernel(
    const float* __restrict__ a, const float* __restrict__ hv_in,
    const float* __restrict__ Wih, const float* __restrict__ Whh,
    const float* __restrict__ bih, const float* __restrict__ bhh,
    float* __restrict__ hv_out)
{
  __shared__ float sG[GRU_WAVES * 2 * 16 * SG_STRIDE];
  const int tid = threadIdx.x, lane = tid & 31, wave = tid >> 5;
  const int hi = lane >> 4, lr = lane & 15;
  float* sGi = &sG[wave * 2 * 16 * SG_STRIDE];
  float* sGh = sGi + 16 * SG_STRIDE;

  const int nTiles     = NNODES / 16;               // 1250, exact
  const int waveGid    = blockIdx.x * GRU_WAVES + wave;
  const int waveStride = gridDim.x * GRU_WAVES;

  const v8f vzero = {0.f,0.f,0.f,0.f,0.f,0.f,0.f,0.f};

  for (int tile = waveGid; tile < nTiles; tile += waveStride) {
    const int nbase = tile * 16;
    v8f acc[12];

    // gi[n,j] = sum_k a[n,k] * Wih[j,k]  -> B frag is contiguous per lane.
#pragma unroll
    for (int nt = 0; nt < 12; ++nt) acc[nt] = vzero;
    for (int kk = 0; kk < NMSG / 4; ++kk) {          // 32 K-steps
      const int kofs = kk * 4 + hi * 2;
      v2f afrag = *(const v2f*)(a + (nbase + lr) * NMSG + kofs);
#pragma unroll
      for (int nt = 0; nt < 12; ++nt) {
        v2f bfrag = *(const v2f*)(Wih + (nt * 16 + lr) * NMSG + kofs);
        acc[nt] = wmma4_f32(afrag, bfrag, acc[nt]);
      }
    }
#pragma unroll
    for (int nt = 0; nt < 12; ++nt) {
      const float bias = bih[nt * 16 + lr];
#pragma unroll
      for (int r = 0; r < 8; ++r)
        sGi[(r + hi * 8) * SG_STRIDE + nt * 16 + lr] = acc[nt][r] + bias;
    }

    // gh[n,j] = sum_k hv[n,k] * Whh[j,k]
#pragma unroll
    for (int nt = 0; nt < 12; ++nt) acc[nt] = vzero;
    for (int kk = 0; kk < HDIM / 4; ++kk) {          // 16 K-steps
      const int kofs = kk * 4 + hi * 2;
      v2f afrag = *(const v2f*)(hv_in + (nbase + lr) * HDIM + kofs);
#pragma unroll
      for (int nt = 0; nt < 12; ++nt) {
        v2f bfrag = *(const v2f*)(Whh + (nt * 16 + lr) * HDIM + kofs);
        acc[nt] = wmma4_f32(afrag, bfrag, acc[nt]);
      }
    }
#pragma unroll
    for (int nt = 0; nt < 12; ++nt) {
      const float bias = bhh[nt * 16 + lr];
#pragma unroll
      for (int r = 0; r < 8; ++r)
        sGh[(r + hi * 8) * SG_STRIDE + nt * 16 + lr] = acc[nt][r] + bias;
    }
    asm volatile("s_wait_dscnt 0" ::: "memory");

    // Gate math: 16 nodes x 64 cols = 1024 elems, 32 per-lane iterations.
#pragma unroll 4
    for (int it = 0; it < 32; ++it) {
      const int idx = it * 32 + lane;
      const int m = idx >> 6, c = idx & 63;
      const float ir  = sGi[m * SG_STRIDE + c];
      const float iz  = sGi[m * SG_STRIDE + 64 + c];
      const float inn = sGi[m * SG_STRIDE + 128 + c];
      const float hr  = sGh[m * SG_STRIDE + c];
      const float hz  = sGh[m * SG_STRIDE + 64 + c];
      const float hn  = sGh[m * SG_STRIDE + 128 + c];
      const float hprev = hv_in[(nbase + m) * HDIM + c];
      const float r = 1.0f / (1.0f + __expf(-(ir + hr)));
      const float z = 1.0f / (1.0f + __expf(-(iz + hz)));
      const float nx = inn + r * hn;
      const float n = 1.0f - 2.0f / (__expf(2.0f * nx) + 1.0f);  // tanh(nx)
      hv_out[(nbase + m) * HDIM + c] = (1.0f - z) * n + z * hprev;
    }
    asm volatile("" ::: "memory");
  }
}

// ---------------------------------------------------------------------------
extern "C" void kernel_launch(void* const* d_in, const int* in_sizes, int n_in,
                              void* d_out, int out_size, void* d_ws, size_t ws_size,
                              hipStream_t stream) {
  const float* hv    = (const float*)d_in[0];
  const float* e     = (const float*)d_in[1];
  const int*   src   = (const int*)d_in[2];
  const int*   dst   = (const int*)d_in[3];
  const float* W_msg = (const float*)d_in[4];  // [2,144,128]
  const float* b_msg = (const float*)d_in[5];  // [2,128]
  const float* W_ih  = (const float*)d_in[6];  // [2,192,128]
  const float* W_hh  = (const float*)d_in[7];  // [2,192,64]
  const float* b_ih  = (const float*)d_in[8];  // [2,192]
  const float* b_hh  = (const float*)d_in[9];  // [2,192]
  float* out = (float*)d_out;

  float* a   = (float*)d_ws;                      // [20000,128] = 10.24 MB
  float* hv1 = a + (size_t)NNODES * NMSG;         // [20000,64]  =  5.12 MB

  const float* hin[2]  = { hv,  hv1 };
  float*       hout[2] = { hv1, out };

  for (int t = 0; t < 2; ++t) {
    zero_kernel<<<2048, 256, 0, stream>>>(a, NNODES * NMSG);
    edge_msg_kernel<<<1024, 128, 0, stream>>>(
        hin[t], e, src, dst,
        W_msg + (size_t)t * KMSG * NMSG, b_msg + t * NMSG, a);
    gru_kernel<<<313, 128, 0, stream>>>(
        a, hin[t],
        W_ih + (size_t)t * H3 * NMSG, W_hh + (size_t)t * H3 * HDIM,
        b_ih + t * H3, b_hh + t * H3, hout[t]);
  }
}